// Hextrema_61615600828978
// MI455X (gfx1250) — compile-verified
//
#include <hip/hip_runtime.h>

#ifndef __has_builtin
#define __has_builtin(x) 0
#endif

#define NEGINF (-1.0e30f)

#if __has_builtin(__builtin_amdgcn_global_load_async_to_lds_b128)
#define HAVE_ASYNC_LD 1
#else
#define HAVE_ASYNC_LD 0
#endif

#if __has_builtin(__builtin_amdgcn_global_store_async_from_lds_b128)
#define HAVE_ASYNC_ST 1
#else
#define HAVE_ASYNC_ST 0
#endif

// Probe-confirmed builtin signature:
//   (v4i __device__* global, v4i __shared__* lds, imm int offset, imm int cpol)
typedef int v4i __attribute__((vector_size(16)));
typedef __attribute__((address_space(1))) v4i* gv4p;   // global int4*
typedef __attribute__((address_space(3))) v4i* lv4p;   // LDS int4*

__device__ __forceinline__ void wait_async0() {
#if __has_builtin(__builtin_amdgcn_s_wait_asynccnt)
    __builtin_amdgcn_s_wait_asynccnt(0);
#else
    asm volatile("s_wait_asynccnt 0" ::: "memory");
#endif
}

__device__ __forceinline__ float vmax3(float u, float m, float d) {
    return fmaxf(fmaxf(u, m), d);
}

// One workgroup per 256x256 image; marker resident in LDS (256 KB of 320 KB).
// Each sweep runs as two 128-column strip phases to keep the barrier-crossing
// register array at 32 floats/thread (no spills). Old column 127 is stashed in
// a 1 KB LDS side buffer so strip 1 still sees Jacobi (old) halo values.
// Thread (ty, tx): row ty (0..255), 32-col sub-segment tx (0..3) of each strip.
__global__ __launch_bounds__(1024)
void hmaxima_kernel(const float* __restrict__ img,
                    const float* __restrict__ hin,
                    float* __restrict__ out)
{
    extern __shared__ __align__(16) float smem[];   // 65536 marker + 256 side
    float* side = smem + 65536;

    const int b   = blockIdx.x;
    const int tid = threadIdx.x;
    const int ty  = tid >> 2;        // 0..255
    const int tx  = tid & 3;         // 0..3

    const float* gimg = img + (size_t)b * 65536u;
    float*       gout = out + (size_t)b * 65536u;
    const float  hb   = hin[b];

    // ---------------- init: img -> LDS (async), marker = img - h ---------
#if HAVE_ASYNC_LD
    {
        #pragma unroll 4
        for (int k = 0; k < 16; ++k) {
            const int byteoff = (tid + (k << 10)) << 4;   // float4-coalesced
            __builtin_amdgcn_global_load_async_to_lds_b128(
                (gv4p)((const char*)gimg + byteoff),
                (lv4p)((char*)smem + byteoff),
                0, 0);
        }
        wait_async0();
    }
    __syncthreads();
    {
        float4* s4 = (float4*)smem;
        #pragma unroll 4
        for (int k = 0; k < 16; ++k) {
            const int i4 = tid + (k << 10);
            float4 v = s4[i4];
            v.x -= hb; v.y -= hb; v.z -= hb; v.w -= hb;
            s4[i4] = v;
        }
    }
    __syncthreads();
#else
    {
        const float4* g4 = (const float4*)gimg;
        float4*       s4 = (float4*)smem;
        #pragma unroll 4
        for (int k = 0; k < 16; ++k) {
            const int i4 = tid + (k << 10);
            float4 v = g4[i4];
            v.x -= hb; v.y -= hb; v.z -= hb; v.w -= hb;
            s4[i4] = v;
        }
    }
    __syncthreads();
#endif

    // ---------------- 128 Jacobi sweeps: m = min(dilate3x3(m), img) ------
    const int  rowbase = ty * 256;
    const bool hasU = (ty > 0);
    const bool hasD = (ty < 255);
    const float4 NEG4 = make_float4(NEGINF, NEGINF, NEGINF, NEGINF);

    for (int it = 0; it < 128; ++it) {
        #pragma unroll
        for (int s = 0; s < 2; ++s) {
            const int c0   = s * 128 + (tx << 5);  // first owned col this strip
            const int base = rowbase + c0;
            const int b4   = base >> 2;
            const float4* s4   = (const float4*)smem;
            const float4* gm4s = (const float4*)gimg + b4;   // mask (L2-resident)

            // vertical max at col c0-1 (left halo / carry seed)
            float prev;
            if (tx == 0) {
                if (s == 0) {
                    prev = NEGINF;                 // image border
                } else {
                    // old col 127 comes from the side buffer (main already new)
                    float m = side[ty];
                    float u = hasU ? side[ty - 1] : NEGINF;
                    float d = hasD ? side[ty + 1] : NEGINF;
                    prev = vmax3(u, m, d);
                }
            } else {
                const int c = base - 1;
                float m = smem[c];
                float u = hasU ? smem[c - 256] : NEGINF;
                float d = hasD ? smem[c + 256] : NEGINF;
                prev = vmax3(u, m, d);
            }

            // rolling vertical-max window: vmN = chunk 0
            float4 vmN;
            {
                float4 m = s4[b4];
                float4 u = hasU ? s4[b4 - 64] : NEG4;
                float4 d = hasD ? s4[b4 + 64] : NEG4;
                vmN.x = vmax3(u.x, m.x, d.x);
                vmN.y = vmax3(u.y, m.y, d.y);
                vmN.z = vmax3(u.z, m.z, d.z);
                vmN.w = vmax3(u.w, m.w, d.w);
            }

            float4 nv[8];
            #pragma unroll
            for (int k = 0; k < 8; ++k) {
                const float4 vmC = vmN;
                float nxt;
                if (k < 7) {
                    const int ci = b4 + k + 1;
                    float4 m = s4[ci];
                    float4 u = hasU ? s4[ci - 64] : NEG4;
                    float4 d = hasD ? s4[ci + 64] : NEG4;
                    vmN.x = vmax3(u.x, m.x, d.x);
                    vmN.y = vmax3(u.y, m.y, d.y);
                    vmN.z = vmax3(u.z, m.z, d.z);
                    vmN.w = vmax3(u.w, m.w, d.w);
                    nxt = vmN.x;
                } else {
                    const int colNext = c0 + 32;           // right halo column
                    if (colNext < 256) {
                        const int c = rowbase + colNext;   // still old in main
                        float m = smem[c];
                        float u = hasU ? smem[c - 256] : NEGINF;
                        float d = hasD ? smem[c + 256] : NEGINF;
                        nxt = vmax3(u, m, d);
                    } else {
                        nxt = NEGINF;                      // image border
                    }
                }

                const float4 mk = gm4s[k];
                float4 r;
                r.x = fminf(vmax3(prev,  vmC.x, vmC.y), mk.x);
                r.y = fminf(vmax3(vmC.x, vmC.y, vmC.z), mk.y);
                r.z = fminf(vmax3(vmC.y, vmC.z, vmC.w), mk.z);
                r.w = fminf(vmax3(vmC.z, vmC.w, nxt),   mk.w);
                prev  = vmC.w;
                nv[k] = r;
            }

            // stash OLD col 127 for strip 1's left halo (read phase => still old)
            if (s == 0 && tx == 3) side[ty] = smem[rowbase + 127];

            __syncthreads();                 // all old-value reads complete
            {
                float4* s4w = (float4*)smem;
                #pragma unroll
                for (int k = 0; k < 8; ++k) s4w[b4 + k] = nv[k];
            }
            __syncthreads();                 // writes visible for next phase
        }
    }

    // ---------------- LDS -> out (async store from LDS) ------------------
#if HAVE_ASYNC_ST
    {
        #pragma unroll 4
        for (int k = 0; k < 16; ++k) {
            const int byteoff = (tid + (k << 10)) << 4;
            __builtin_amdgcn_global_store_async_from_lds_b128(
                (gv4p)((char*)gout + byteoff),
                (lv4p)((char*)smem + byteoff),
                0, 0);
        }
        wait_async0();
    }
#else
    {
        const float4* s4 = (const float4*)smem;
        float4*       g4 = (float4*)gout;
        #pragma unroll 4
        for (int k = 0; k < 16; ++k) {
            const int i4 = tid + (k << 10);
            g4[i4] = s4[i4];
        }
    }
#endif
}

extern "C" void kernel_launch(void* const* d_in, const int* in_sizes, int n_in,
                              void* d_out, int out_size, void* d_ws, size_t ws_size,
                              hipStream_t stream) {
    const float* img = (const float*)d_in[0];
    const float* h   = (const float*)d_in[1];
    float*       out = (float*)d_out;

    const int B = in_sizes[0] / (256 * 256);
    const size_t lds_bytes = (256 * 256 + 256) * sizeof(float);  // marker + side

    // Allow >64KB dynamic LDS (no stream work; safe under graph capture).
    (void)hipFuncSetAttribute((const void*)hmaxima_kernel,
                              hipFuncAttributeMaxDynamicSharedMemorySize,
                              (int)lds_bytes);

    hmaxima_kernel<<<B, 1024, lds_bytes, stream>>>(img, h, out);
}